// BaseBlock_21534966022853
// MI455X (gfx1250) — compile-verified
//
#include <hip/hip_runtime.h>

// MI455X / gfx1250 implementation of LocalGrouper (FPS + kNN + anchor gather).
//
// Perf model (MI455X, 23.3 TB/s HBM):
//  - FLOPs negligible (~1.6 GFLOP distance GEMM, K=3) -> latency/BW bound.
//  - Dominant traffic: final gather ~1.1 GB -> ~50us at HBM roofline; gather
//    kernel is pure b128 load/store streams.
//  - square_distance einsum runs on V_WMMA_F32_16X16X4_F32, folding -2*q.p
//    and +|p|^2 into one K=4 WMMA (per-row |q|^2 constant dropped: cannot
//    change per-row top-k ordering). Two WMMAs per B tile (32 query rows per
//    wave) so all 32 lanes own a row during top-k selection.
//  - B tiles are software-pipelined one deep to hide global load latency
//    behind WMMA + LDS selection.
//  - FPS is the serial part: persistent block per batch, points in registers,
//    wave32 shuffle argmax + 32-entry LDS cross-wave reduction.

#define BATCH 32
#define NPTS  8192
#define NGRP  1024
#define KNN_K 32
#define DIMF  128

typedef __attribute__((ext_vector_type(2))) float v2f;
typedef __attribute__((ext_vector_type(8))) float v8f;

// ---------------------------------------------------------------- K1: pack B
// pb[b][n] = (x, y, z, |p|^2) so one WMMA computes -2 q.p + |p|^2.
__global__ __launch_bounds__(256) void prep_kernel(const float* __restrict__ xyz,
                                                   float4* __restrict__ pb) {
    int i = blockIdx.x * blockDim.x + threadIdx.x;
    if (i >= BATCH * NPTS) return;
    float x = xyz[i * 3 + 0], y = xyz[i * 3 + 1], z = xyz[i * 3 + 2];
    pb[i] = make_float4(x, y, z, x * x + y * y + z * z);
}

// ---------------------------------------------------------------- K2: FPS
// One block per batch, 1024 threads (32 wave32 waves). Each thread owns 8
// points in registers. 1024 sequential argmax steps; per step: wave-level
// shuffle reduction then a single 32-entry cross-wave reduction.
__global__ __launch_bounds__(1024) void fps_kernel(const float* __restrict__ xyz,
                                                   int* __restrict__ fpsidx,
                                                   float* __restrict__ out_xyz) {
    const int b = blockIdx.x;
    const int t = threadIdx.x;
    const float* bx = xyz + (size_t)b * NPTS * 3;

    float px[8], py[8], pz[8], dist[8];
#pragma unroll
    for (int i = 0; i < 8; ++i) {
        int n = i * 1024 + t;                 // strided ownership, coalesced load
        px[i] = bx[n * 3 + 0];
        py[i] = bx[n * 3 + 1];
        pz[i] = bx[n * 3 + 2];
        dist[i] = 1e10f;
    }

    __shared__ float wv[32];
    __shared__ int   wi[32];
    __shared__ float s_cx, s_cy, s_cz;
    __shared__ int   s_cur;
    if (t == 0) s_cur = 0;                    // deterministic start at index 0

    const int wave = t >> 5;
    const int lane = t & 31;

    for (int it = 0; it < NGRP; ++it) {
        __syncthreads();                      // s_cur from previous step visible
        int cur = s_cur;
        if ((cur & 1023) == t) {              // owner thread broadcasts centroid
            int i = cur >> 10;
            s_cx = px[i]; s_cy = py[i]; s_cz = pz[i];
        }
        if (t == 0) fpsidx[b * NGRP + it] = cur;   // scan emits OLD farthest
        __syncthreads();

        float cx = s_cx, cy = s_cy, cz = s_cz;
        if (t == 0) {                         // new_xyz output (first tuple elem)
            float* o = out_xyz + ((size_t)b * NGRP + it) * 3;
            o[0] = cx; o[1] = cy; o[2] = cz;
        }

        float bv = -1.0f; int bi = 0;
#pragma unroll
        for (int i = 0; i < 8; ++i) {
            float dx = px[i] - cx, dy = py[i] - cy, dz = pz[i] - cz;
            float dd = dx * dx + dy * dy + dz * dz;
            float nd = fminf(dist[i], dd);
            dist[i] = nd;
            if (nd > bv) { bv = nd; bi = i * 1024 + t; }
        }
        // wave32 argmax reduction
        for (int off = 16; off > 0; off >>= 1) {
            float ov = __shfl_down(bv, off);
            int   oi = __shfl_down(bi, off);
            if (ov > bv) { bv = ov; bi = oi; }
        }
        if (lane == 0) { wv[wave] = bv; wi[wave] = bi; }
        __syncthreads();
        if (t < 32) {                          // wave 0 reduces the 32 wave maxima
            bv = wv[t]; bi = wi[t];
            for (int off = 16; off > 0; off >>= 1) {
                float ov = __shfl_down(bv, off);
                int   oi = __shfl_down(bi, off);
                if (ov > bv) { bv = ov; bi = oi; }
            }
            if (t == 0) s_cur = bi;
        }
    }
}

// ---------------------------------------------------------------- K3: kNN
// One wave per (batch, 32-query tile). Distances via V_WMMA_F32_16X16X4_F32:
//   A (16x4): row m = (-2qx, -2qy, -2qz, 1)   B (4x16): col n = (px,py,pz,|p|^2)
//   score = -2 q.p + |p|^2   (|q|^2 per-row constant -> ordering unchanged)
// Two WMMAs per B tile (rows 0-15 and 16-31); C layout: VGPR r holds row r
// (lanes 0-15) / row r+8 (lanes 16-31), col = lane%16. All 32 lanes then run
// the per-row sorted top-32 insertion list.
__global__ __launch_bounds__(32) void knn_kernel(const float* __restrict__ xyz,
                                                 const float4* __restrict__ pb,
                                                 const int* __restrict__ fpsidx,
                                                 int* __restrict__ knnidx) {
    const int tile  = blockIdx.x;         // 0..31  (32 queries per tile)
    const int b     = blockIdx.y;         // 0..31  (batch)
    const int lane  = threadIdx.x;        // wave32
    const int m     = lane & 15;
    const int sbase = tile * 32;

    // Each lane loads query row m of both 16-row sub-tiles; component choice
    // by lane half matches the 32-bit 16x4 A layout (lanes 0-15: K0,K1;
    // lanes 16-31: K2,K3).
    int qi0 = fpsidx[b * NGRP + sbase + m];
    int qi1 = fpsidx[b * NGRP + sbase + 16 + m];
    const float* q0 = xyz + ((size_t)b * NPTS + qi0) * 3;
    const float* q1 = xyz + ((size_t)b * NPTS + qi1) * 3;
    v2f a0, a1;
    if (lane < 16) {
        a0.x = -2.0f * q0[0]; a0.y = -2.0f * q0[1];
        a1.x = -2.0f * q1[0]; a1.y = -2.0f * q1[1];
    } else {
        a0.x = -2.0f * q0[2]; a0.y = 1.0f;
        a1.x = -2.0f * q1[2]; a1.y = 1.0f;
    }

    __shared__ float s_sc[32 * 16];
    __shared__ float s_tv[32 * KNN_K];
    __shared__ int   s_ti[32 * KNN_K];
    for (int j = lane; j < 32 * KNN_K; j += 32) { s_tv[j] = 3.4e38f; s_ti[j] = 0; }
    __syncthreads();

    const float4* pbb   = pb + (size_t)b * NPTS;
    const int     rowoff = (lane >> 4) << 3;

    float4 p = pbb[m];                        // pipeline: tile 0 in flight
    for (int ct = 0; ct < NPTS / 16; ++ct) {
        float4 pc = p;
        if (ct + 1 < NPTS / 16)               // prefetch next B tile; its load
            p = pbb[(ct + 1) * 16 + m];       // latency hides under WMMA+select

        // B: 4x16 — lanes 0-15 supply K0,K1 ; lanes 16-31 supply K2,K3
        v2f bb;
        if (lane < 16) { bb.x = pc.x; bb.y = pc.y; }
        else           { bb.x = pc.z; bb.y = pc.w; }

        v8f c0 = {}, c1 = {};
        c0 = __builtin_amdgcn_wmma_f32_16x16x4_f32(
                false, a0, false, bb, (short)0, c0, false, false);
        c1 = __builtin_amdgcn_wmma_f32_16x16x4_f32(
                false, a1, false, bb, (short)0, c1, false, false);

#pragma unroll
        for (int r = 0; r < 8; ++r) {
            s_sc[(r + rowoff) * 16 + m]        = c0[r];
            s_sc[(16 + r + rowoff) * 16 + m]   = c1[r];
        }
        __syncthreads();

        // every lane owns one of the 32 query rows
        {
            float* tv = s_tv + lane * KNN_K;
            int*   ti = s_ti + lane * KNN_K;
            const float* sc_row = s_sc + lane * 16;
            for (int j = 0; j < 16; ++j) {
                float sc = sc_row[j];
                if (sc < tv[KNN_K - 1]) {
                    int p2 = KNN_K - 1;
                    while (p2 > 0 && tv[p2 - 1] > sc) {
                        tv[p2] = tv[p2 - 1]; ti[p2] = ti[p2 - 1]; --p2;
                    }
                    tv[p2] = sc; ti[p2] = ct * 16 + j;
                }
            }
        }
        __syncthreads();
    }

    int* o = knnidx + ((size_t)b * NGRP + sbase + lane) * KNN_K;
    for (int k = 0; k < KNN_K; ++k) o[k] = s_ti[lane * KNN_K + k];
}

// ---------------------------------------------------------------- K4: gather
// a[b][s][k][d] = points[b][knn[k]][d] - points[b][fps[s]][d]
// Block per (b,s); 256 threads, float4 lanes: 8 neighbor rows in flight,
// b128 coalesced loads/stores (this kernel carries the HBM roofline traffic).
__global__ __launch_bounds__(256) void gather_kernel(const float* __restrict__ points,
                                                     const int* __restrict__ fpsidx,
                                                     const int* __restrict__ knnidx,
                                                     float* __restrict__ out_a) {
    int bs = blockIdx.x;                 // 0..B*S-1
    int b  = bs >> 10;
    int s  = bs & 1023;
    int t  = threadIdx.x;
    int d4 = t & 31;                     // float4 index within 128-float row
    int kg = t >> 5;                     // k-group 0..7

    const float4* bp = (const float4*)(points + (size_t)b * NPTS * DIMF);
    int qi = fpsidx[b * NGRP + s];
    float4 anchor = bp[(size_t)qi * (DIMF / 4) + d4];

    const int* ki = knnidx + (size_t)bs * KNN_K;
    float4* ob = (float4*)(out_a + (size_t)bs * KNN_K * DIMF);

#pragma unroll
    for (int k = kg; k < KNN_K; k += 8) {
        int n = ki[k];
        float4 v = bp[(size_t)n * (DIMF / 4) + d4];
        ob[k * (DIMF / 4) + d4] = make_float4(v.x - anchor.x, v.y - anchor.y,
                                              v.z - anchor.z, v.w - anchor.w);
    }
}

// ---------------------------------------------------------------- launch
extern "C" void kernel_launch(void* const* d_in, const int* in_sizes, int n_in,
                              void* d_out, int out_size, void* d_ws, size_t ws_size,
                              hipStream_t stream) {
    (void)in_sizes; (void)n_in; (void)out_size; (void)ws_size;
    const float* xyz    = (const float*)d_in[0];   // [32,8192,3]
    const float* points = (const float*)d_in[1];   // [32,8192,128]

    float* out      = (float*)d_out;
    float* out_xyz  = out;                                   // [32,1024,3]
    float* out_a    = out + (size_t)BATCH * NGRP * 3;        // [32,1024,32,128]

    // workspace layout (bytes): pb 4MB | fpsidx 128KB | knnidx 4MB
    float4* pb   = (float4*)d_ws;
    int* fpsidx  = (int*)((char*)d_ws + (size_t)BATCH * NPTS * sizeof(float4));
    int* knnidx  = (int*)((char*)fpsidx + (size_t)BATCH * NGRP * sizeof(int));

    prep_kernel<<<(BATCH * NPTS) / 256, 256, 0, stream>>>(xyz, pb);
    fps_kernel<<<BATCH, 1024, 0, stream>>>(xyz, fpsidx, out_xyz);
    knn_kernel<<<dim3(NGRP / 32, BATCH), 32, 0, stream>>>(xyz, pb, fpsidx, knnidx);
    gather_kernel<<<BATCH * NGRP, 256, 0, stream>>>(points, fpsidx, knnidx, out_a);
}